// TorchVQC_15290083573821
// MI455X (gfx1250) — compile-verified
//
#include <hip/hip_runtime.h>

// fp32 WMMA fragments for V_WMMA_F32_16X16X4_F32 (wave32):
//  A 16x4  f32 -> float2/lane: lane m=L&15, K = 2*(L>>4) + {0,1}
//  B 4x16  f32 -> float2/lane: lane n=L&15, K = 2*(L>>4) + {0,1}
//  C/D 16x16 f32 -> 8 VGPRs: reg r holds M = r + 8*(L>>4), N = L&15
typedef float v2f __attribute__((ext_vector_type(2)));
typedef float v8f __attribute__((ext_vector_type(8)));

#define WMMA_F32_16x16x4(a, b, c) \
  __builtin_amdgcn_wmma_f32_16x16x4_f32(false, (a), false, (b), (short)0, (c), false, false)

// LDS bank swizzle: XOR bank bits [5:2] with g0 (bits 15:12), g1 (bits 11:8)
// and g2-high (bits 7:6). Modifies only bits 5:2, mask depends on bits 15:6
// => involution => bijection. Kills 16/32-way conflicts of the stride-256 /
// stride-4096 phase-B accesses and the stride-16 phase-A accesses.
// Note SW(i+1) == SW(i)+1 for even i, so packed b64 accesses stay legal.
__device__ __forceinline__ int SW(int i) {
    return i ^ ((i >> 10) & 0x3C) ^ ((i >> 6) & 0x3C) ^ ((i >> 4) & 0xC);
}

// 16 qubits, batch 512, depth 6, n_class 4.
// One workgroup per sample; full 2^16-amplitude state lives in LDS (256 KB).
__launch_bounds__(512, 1)
__global__ void vqc16_kernel(const float* __restrict__ X,
                             const float* __restrict__ wts,
                             float* __restrict__ out)
{
    __shared__ __align__(16) float st[65536];      // 256 KB state (swizzled)
    __shared__ __align__(16) float Umat[4][320];   // 16x16 matrices, row stride 20
    __shared__ float csb[16][2];
    __shared__ float afac[16][2];
    __shared__ float gfac[4][16];
    __shared__ float wsum[16];

    const int tid  = threadIdx.x;
    const int blk  = blockIdx.x;
    const int lane = tid & 31;
    const int wave = tid >> 5;          // 16 waves; wave == g0 block for readout
    const int h    = lane >> 4;         // half-wave
    const int mn   = lane & 15;         // row (A) / col (B,D) index
    const int k0   = 2 * h;             // K offset within a K=4 chunk

    // ---- Initial product state: RY(X[b,q]) on the uniform state ----
    if (tid < 16) {
        float x = 0.5f * X[blk * 16 + tid];
        float c = cosf(x), s = sinf(x);
        afac[tid][0] = (c - s) * 0.7071067811865476f;
        afac[tid][1] = (c + s) * 0.7071067811865476f;
    }
    __syncthreads();
    if (tid < 64) {
        int g = tid >> 4, v = tid & 15;
        float f = 1.0f;
        #pragma unroll
        for (int t = 0; t < 4; ++t) f *= afac[4 * g + t][(v >> (3 - t)) & 1];
        gfac[g][v] = f;
    }
    __syncthreads();
    {
        int base = tid * 128;
        for (int k = 0; k < 128; ++k) {
            int i = base + k;
            st[SW(i)] = gfac[0][(i >> 12) & 15] * gfac[1][(i >> 8) & 15]
                      * gfac[2][(i >> 4) & 15] * gfac[3][i & 15];
        }
    }
    __syncthreads();

    for (int layer = 0; layer < 6; ++layer) {
        // ---- CNOT even sublayer: ctrl bits 15,13,...,1 (mask 0xAAAA) ----
        {
            int base = tid * 128;
            for (int k = 0; k < 128; ++k) {
                int i = base + k;
                int j = i ^ ((i & 0xAAAA) >> 1);
                if (j > i) {
                    int pi = SW(i), pj = SW(j);
                    float t0 = st[pi]; st[pi] = st[pj]; st[pj] = t0;
                }
            }
        }
        __syncthreads();
        // ---- CNOT odd sublayer: ctrl bits 14,12,...,2 (mask 0x5554) ----
        {
            int base = tid * 128;
            for (int k = 0; k < 128; ++k) {
                int i = base + k;
                int j = i ^ ((i & 0x5554) >> 1);
                if (j > i) {
                    int pi = SW(i), pj = SW(j);
                    float t0 = st[pi]; st[pi] = st[pj]; st[pj] = t0;
                }
            }
        }
        __syncthreads();

        // ---- Build the four 16x16 group rotation matrices for this layer ----
        if (tid < 16) {
            float x = 0.5f * wts[layer * 16 + tid];
            csb[tid][0] = cosf(x);
            csb[tid][1] = sinf(x);
        }
        __syncthreads();
        for (int idx = tid; idx < 1024; idx += 512) {
            int g = idx >> 8, rr = (idx >> 4) & 15, cc = idx & 15;
            float v = 1.0f;
            #pragma unroll
            for (int t = 0; t < 4; ++t) {
                float c = csb[4 * g + t][0], s = csb[4 * g + t][1];
                int rb = (rr >> (3 - t)) & 1, cb = (cc >> (3 - t)) & 1;
                v *= rb ? (cb ? c : s) : (cb ? -s : c);
            }
            Umat[g][rr * 20 + cc] = v;
        }
        __syncthreads();

        // ---- Phase A: contiguous tiles (g0,g1): M' = U2 * (M * U3^T) ----
        // Two independent tiles per iteration for ILP on the WMMA chains.
        for (int tt = 0; tt < 16; tt += 2) {
            int tb0 = (wave * 16 + tt) << 8;
            int tb1 = tb0 + 256;

            v8f w0 = {0.f,0.f,0.f,0.f,0.f,0.f,0.f,0.f};
            v8f w1 = {0.f,0.f,0.f,0.f,0.f,0.f,0.f,0.f};
            #pragma unroll
            for (int c = 0; c < 4; ++c) {          // W = M * U3^T
                v2f a0 = *(const v2f*)&st[SW(tb0 + mn * 16 + 4 * c + k0)];
                v2f a1 = *(const v2f*)&st[SW(tb1 + mn * 16 + 4 * c + k0)];
                v2f bv = *(const v2f*)&Umat[3][mn * 20 + 4 * c + k0];
                w0 = WMMA_F32_16x16x4(a0, bv, w0);
                w1 = WMMA_F32_16x16x4(a1, bv, w1);
            }
            #pragma unroll
            for (int r = 0; r < 8; ++r) {          // park W in-place
                st[SW(tb0 + (r + 8 * h) * 16 + mn)] = w0[r];
                st[SW(tb1 + (r + 8 * h) * 16 + mn)] = w1[r];
            }
            v8f d0 = {0.f,0.f,0.f,0.f,0.f,0.f,0.f,0.f};
            v8f d1 = {0.f,0.f,0.f,0.f,0.f,0.f,0.f,0.f};
            #pragma unroll
            for (int c = 0; c < 4; ++c) {          // D = U2 * W
                v2f a = *(const v2f*)&Umat[2][mn * 20 + 4 * c + k0];
                v2f b0, b1;
                b0.x = st[SW(tb0 + (4 * c + k0) * 16 + mn)];
                b0.y = st[SW(tb0 + (4 * c + k0 + 1) * 16 + mn)];
                b1.x = st[SW(tb1 + (4 * c + k0) * 16 + mn)];
                b1.y = st[SW(tb1 + (4 * c + k0 + 1) * 16 + mn)];
                d0 = WMMA_F32_16x16x4(a, b0, d0);
                d1 = WMMA_F32_16x16x4(a, b1, d1);
            }
            #pragma unroll
            for (int r = 0; r < 8; ++r) {
                st[SW(tb0 + (r + 8 * h) * 16 + mn)] = d0[r];
                st[SW(tb1 + (r + 8 * h) * 16 + mn)] = d1[r];
            }
        }
        __syncthreads();

        // ---- Phase B: strided tiles (g2,g3): M' = U0 * (M01 * U1^T) ----
        // Tile pair (g23, g23+1) -> every state access is a packed b64.
        for (int tt = 0; tt < 16; tt += 2) {
            int g23 = wave * 16 + tt;              // even

            v8f w0 = {0.f,0.f,0.f,0.f,0.f,0.f,0.f,0.f};
            v8f w1 = {0.f,0.f,0.f,0.f,0.f,0.f,0.f,0.f};
            #pragma unroll
            for (int c = 0; c < 4; ++c) {          // W = M01 * U1^T
                int col = 4 * c + k0;
                v2f p0 = *(const v2f*)&st[SW(mn * 4096 + col * 256 + g23)];
                v2f p1 = *(const v2f*)&st[SW(mn * 4096 + (col + 1) * 256 + g23)];
                v2f a0 = {p0.x, p1.x};             // tile g23
                v2f a1 = {p0.y, p1.y};             // tile g23+1
                v2f bv = *(const v2f*)&Umat[1][mn * 20 + 4 * c + k0];
                w0 = WMMA_F32_16x16x4(a0, bv, w0);
                w1 = WMMA_F32_16x16x4(a1, bv, w1);
            }
            #pragma unroll
            for (int r = 0; r < 8; ++r) {
                v2f pw = {w0[r], w1[r]};
                *(v2f*)&st[SW((r + 8 * h) * 4096 + mn * 256 + g23)] = pw;
            }
            v8f d0 = {0.f,0.f,0.f,0.f,0.f,0.f,0.f,0.f};
            v8f d1 = {0.f,0.f,0.f,0.f,0.f,0.f,0.f,0.f};
            #pragma unroll
            for (int c = 0; c < 4; ++c) {          // D = U0 * W
                v2f a  = *(const v2f*)&Umat[0][mn * 20 + 4 * c + k0];
                v2f q0 = *(const v2f*)&st[SW((4 * c + k0) * 4096 + mn * 256 + g23)];
                v2f q1 = *(const v2f*)&st[SW((4 * c + k0 + 1) * 4096 + mn * 256 + g23)];
                v2f b0 = {q0.x, q1.x};
                v2f b1 = {q0.y, q1.y};
                d0 = WMMA_F32_16x16x4(a, b0, d0);
                d1 = WMMA_F32_16x16x4(a, b1, d1);
            }
            #pragma unroll
            for (int r = 0; r < 8; ++r) {
                v2f pd = {d0[r], d1[r]};
                *(v2f*)&st[SW((r + 8 * h) * 4096 + mn * 256 + g23)] = pd;
            }
        }
        __syncthreads();
    }

    // ---- Readout: z_w = sum probs * sign(bit 15-w). The swizzle permutes only
    // within each aligned 128-element chunk and preserves bits 15:12, so a
    // physical-order sweep per thread is exact (signs are wave-uniform). ----
    float partial = 0.f;
    {
        int base = tid * 128;
        for (int k = 0; k < 128; ++k) { float v = st[base + k]; partial += v * v; }
    }
    #pragma unroll
    for (int off = 16; off >= 1; off >>= 1) partial += __shfl_xor(partial, off, 32);
    if (lane == 0) wsum[wave] = partial;
    __syncthreads();
    if (tid < 4) {
        float z = 0.f;
        #pragma unroll
        for (int g0 = 0; g0 < 16; ++g0)
            z += (((g0 >> (3 - tid)) & 1) ? -1.f : 1.f) * wsum[g0];
        out[blk * 4 + tid] = z;
    }
}

extern "C" void kernel_launch(void* const* d_in, const int* in_sizes, int n_in,
                              void* d_out, int out_size, void* d_ws, size_t ws_size,
                              hipStream_t stream) {
    const float* X   = (const float*)d_in[0];   // (512,16) f32
    const float* wts = (const float*)d_in[1];   // (6,16) f32
    float* out = (float*)d_out;                 // (512,4) f32
    int B = in_sizes[0] / 16;
    vqc16_kernel<<<B, 512, 0, stream>>>(X, wts, out);
}